// GroupedExperts_35751307772685
// MI455X (gfx1250) — compile-verified
//
#include <hip/hip_runtime.h>
#include <hip/hip_bf16.h>
#include <math.h>
#include <stdint.h>

// ---------------- problem constants ----------------
#define NE   64
#define HID  2048
#define INTR 768
#define TOK  65536

// ---------------- tiling ----------------
#define TM   128          // workgroup M tile
#define TN   128          // workgroup N tile
#define KC   32           // K chunk staged in LDS
#define LDA  (KC + 4)     // x-tile LDS row stride (36): conflict-free A-frag b64 reads
#define LDB  (TN + 16)    // w-tile LDS row stride (144): halves offset by 2*144 ≡ 32 banks
#define XSZ  (TM * LDA)   // 4608 floats per x-tile buffer
#define WSZ  (KC * LDB)   // 4608 floats per weight-tile buffer
#define MAX_MT 16         // covers up to 2048 tokens per expert (actual max is 1280)

typedef float v2f __attribute__((ext_vector_type(2)));
typedef float v8f __attribute__((ext_vector_type(8)));

__device__ __forceinline__ v8f wmma_f32x4(v2f a, v2f b, v8f c) {
  // D = A(16x4 f32) * B(4x16 f32) + C(16x16 f32)
  return __builtin_amdgcn_wmma_f32_16x16x4_f32(
      /*neg_a=*/false, a, /*neg_b=*/false, b,
      /*c_mod=*/(short)0, c, /*reuse_a=*/false, /*reuse_b=*/false);
}

// Async 16B copy global -> LDS (ASYNCcnt-tracked, no VGPR round trip).
// Generic pointers to LDS carry the AS(3) byte offset in their low 32 bits.
__device__ __forceinline__ void async_b128(void* lds, const float* g) {
  uint32_t off = (uint32_t)(uintptr_t)lds;
  asm volatile("global_load_async_to_lds_b128 %0, %1, off"
               :: "v"(off), "v"(g) : "memory");
}

// ------------------------------------------------------------------
// Kernel 0: expert token-offset prefix sum (counts live on device)
// ------------------------------------------------------------------
__global__ void moe_prefix_kernel(const int* __restrict__ counts,
                                  int* __restrict__ offs) {
  if (threadIdx.x == 0) {
    int s = 0;
    for (int i = 0; i < NE; ++i) { offs[i] = s; s += counts[i]; }
    offs[NE] = s;
  }
}

// ------------------------------------------------------------------
// Kernel 1: h = silu(x @ gate[e]) * (x @ up[e])   [c_e, INTR] per expert
// grid: (INTR/TN, MAX_MT, NE), block 256 (8 waves, 2x4)
// dynamic LDS: lx[2][XSZ] | lwg[2][WSZ] | lwu[2][WSZ]  (110592 B)
// ------------------------------------------------------------------
__global__ void __launch_bounds__(256)
moe_gateup_kernel(const float* __restrict__ x,
                  const float* __restrict__ gate,
                  const float* __restrict__ up,
                  const int* __restrict__ offs,
                  float* __restrict__ h) {
  const int nt = blockIdx.x;     // INTER tile
  const int mt = blockIdx.y;     // token tile within expert
  const int e  = blockIdx.z;     // expert

  const int start = offs[e];
  const int c     = offs[e + 1] - start;
  if (mt * TM >= c) return;
  const int mlim = c - mt * TM;  // valid rows in this tile (>=1)

  extern __shared__ float smem[];
  float* lx  = smem;                       // 2 * XSZ
  float* lwg = smem + 2 * XSZ;             // 2 * WSZ
  float* lwu = smem + 2 * XSZ + 2 * WSZ;   // 2 * WSZ

  const int tid  = threadIdx.x;
  const int lane = tid & 31;
  const int wave = tid >> 5;
  const int wm   = wave & 1;        // wave M block (0..1) -> 64 rows
  const int wn   = wave >> 1;       // wave N block (0..3) -> 32 cols
  const int half = lane >> 4;       // lane half (WMMA layout)
  const int lr   = lane & 15;

  const float* gW = gate + (size_t)e * HID * INTR + (size_t)nt * TN;
  const float* uW = up   + (size_t)e * HID * INTR + (size_t)nt * TN;
  const float* xP = x + (size_t)(start + mt * TM) * HID;

  // per-thread staging coordinates
  const int xr0 = tid >> 3;          // x tile: 8 thr/row
  const int xc  = (tid & 7) << 2;
  const int wr0 = tid >> 5;          // w tile: 32 thr/row
  const int wc  = (tid & 31) << 2;

  // stage one K-chunk into buffer `b` (12 async b128 per thread/wave)
  auto stage = [&](int b, int kb) {
    int r = xr0;
    #pragma unroll
    for (int p = 0; p < 4; ++p, r += 32) {
      const int rr = r < mlim ? r : (mlim - 1);   // clamp: uniform issue count
      async_b128(&lx[b * XSZ + r * LDA + xc],
                 xP + (size_t)rr * HID + kb + xc);
    }
    int r2 = wr0;
    #pragma unroll
    for (int p = 0; p < 4; ++p, r2 += 8) {
      async_b128(&lwg[b * WSZ + r2 * LDB + wc],
                 gW + (size_t)(kb + r2) * INTR + wc);
      async_b128(&lwu[b * WSZ + r2 * LDB + wc],
                 uW + (size_t)(kb + r2) * INTR + wc);
    }
  };

  v8f accG[4][2] = {};
  v8f accU[4][2] = {};

  stage(0, 0);
  int buf = 0;
  for (int kb = 0; kb < HID; kb += KC) {
    if (kb + KC < HID) {
      stage(buf ^ 1, kb + KC);
      asm volatile("s_wait_asynccnt 12" ::: "memory");  // chunk kb landed
    } else {
      asm volatile("s_wait_asynccnt 0" ::: "memory");
    }
    __syncthreads();

    const float* bx = &lx[buf * XSZ];
    const float* bg = &lwg[buf * WSZ];
    const float* bu = &lwu[buf * WSZ];

    #pragma unroll
    for (int k = 0; k < KC; k += 4) {
      v2f a[4], fg[2], fu[2];
      #pragma unroll
      for (int i = 0; i < 4; ++i) {
        const int row = wm * 64 + i * 16 + lr;
        a[i] = *(const v2f*)(&bx[row * LDA + k + 2 * half]);
      }
      #pragma unroll
      for (int j = 0; j < 2; ++j) {
        const int col  = wn * 32 + j * 16 + lr;
        const int krow = k + 2 * half;
        fg[j].x = bg[krow * LDB + col];
        fg[j].y = bg[(krow + 1) * LDB + col];
        fu[j].x = bu[krow * LDB + col];
        fu[j].y = bu[(krow + 1) * LDB + col];
      }
      #pragma unroll
      for (int i = 0; i < 4; ++i)
        #pragma unroll
        for (int j = 0; j < 2; ++j) {
          accG[i][j] = wmma_f32x4(a[i], fg[j], accG[i][j]);
          accU[i][j] = wmma_f32x4(a[i], fu[j], accU[i][j]);
        }
    }
    __syncthreads();   // all waves done reading buf before async overwrites it
    buf ^= 1;
  }

  // ---- epilogue: h = g*sigmoid(g)*u, C layout: VGPR r -> rows r / r+8 ----
  #pragma unroll
  for (int i = 0; i < 4; ++i) {
    const int mloc = wm * 64 + i * 16 + half * 8;
    #pragma unroll
    for (int j = 0; j < 2; ++j) {
      const int col = nt * TN + wn * 32 + j * 16 + lr;
      #pragma unroll
      for (int r = 0; r < 8; ++r) {
        const int m = mt * TM + mloc + r;
        if (m < c) {
          const float g = accG[i][j][r];
          const float u = accU[i][j][r];
          const float s = g / (1.0f + __expf(-g));
          h[(size_t)(start + m) * INTR + col] = s * u;
        }
      }
    }
  }
}

// ------------------------------------------------------------------
// Kernel 2: out = h @ down[e]   [c_e, HID] per expert
// grid: (HID/TN, MAX_MT, NE), block 256
// dynamic LDS: lx[2][XSZ] | lw[2][WSZ]  (73728 B)
// ------------------------------------------------------------------
__global__ void __launch_bounds__(256)
moe_down_kernel(const float* __restrict__ h,
                const float* __restrict__ down,
                const int* __restrict__ offs,
                float* __restrict__ out) {
  const int nt = blockIdx.x;
  const int mt = blockIdx.y;
  const int e  = blockIdx.z;

  const int start = offs[e];
  const int c     = offs[e + 1] - start;
  if (mt * TM >= c) return;
  const int mlim = c - mt * TM;

  extern __shared__ float smem[];
  float* lx = smem;              // 2 * XSZ
  float* lw = smem + 2 * XSZ;    // 2 * WSZ

  const int tid  = threadIdx.x;
  const int lane = tid & 31;
  const int wave = tid >> 5;
  const int wm   = wave & 1;
  const int wn   = wave >> 1;
  const int half = lane >> 4;
  const int lr   = lane & 15;

  const float* dW = down + (size_t)e * INTR * HID + (size_t)nt * TN;
  const float* hP = h + (size_t)(start + mt * TM) * INTR;

  const int xr0 = tid >> 3;
  const int xc  = (tid & 7) << 2;
  const int wr0 = tid >> 5;
  const int wc  = (tid & 31) << 2;

  auto stage = [&](int b, int kb) {   // 8 async b128 per thread/wave
    int r = xr0;
    #pragma unroll
    for (int p = 0; p < 4; ++p, r += 32) {
      const int rr = r < mlim ? r : (mlim - 1);
      async_b128(&lx[b * XSZ + r * LDA + xc],
                 hP + (size_t)rr * INTR + kb + xc);
    }
    int r2 = wr0;
    #pragma unroll
    for (int p = 0; p < 4; ++p, r2 += 8) {
      async_b128(&lw[b * WSZ + r2 * LDB + wc],
                 dW + (size_t)(kb + r2) * HID + wc);
    }
  };

  v8f acc[4][2] = {};

  stage(0, 0);
  int buf = 0;
  for (int kb = 0; kb < INTR; kb += KC) {
    if (kb + KC < INTR) {
      stage(buf ^ 1, kb + KC);
      asm volatile("s_wait_asynccnt 8" ::: "memory");
    } else {
      asm volatile("s_wait_asynccnt 0" ::: "memory");
    }
    __syncthreads();

    const float* bx = &lx[buf * XSZ];
    const float* bw = &lw[buf * WSZ];

    #pragma unroll
    for (int k = 0; k < KC; k += 4) {
      v2f a[4], b[2];
      #pragma unroll
      for (int i = 0; i < 4; ++i) {
        const int row = wm * 64 + i * 16 + lr;
        a[i] = *(const v2f*)(&bx[row * LDA + k + 2 * half]);
      }
      #pragma unroll
      for (int j = 0; j < 2; ++j) {
        const int col  = wn * 32 + j * 16 + lr;
        const int krow = k + 2 * half;
        b[j].x = bw[krow * LDB + col];
        b[j].y = bw[(krow + 1) * LDB + col];
      }
      #pragma unroll
      for (int i = 0; i < 4; ++i)
        #pragma unroll
        for (int j = 0; j < 2; ++j)
          acc[i][j] = wmma_f32x4(a[i], b[j], acc[i][j]);
    }
    __syncthreads();
    buf ^= 1;
  }

  #pragma unroll
  for (int i = 0; i < 4; ++i) {
    const int mloc = wm * 64 + i * 16 + half * 8;
    #pragma unroll
    for (int j = 0; j < 2; ++j) {
      const int col = nt * TN + wn * 32 + j * 16 + lr;
      #pragma unroll
      for (int r = 0; r < 8; ++r) {
        const int m = mt * TM + mloc + r;
        if (m < c)
          out[(size_t)(start + m) * HID + col] = acc[i][j][r];
      }
    }
  }
}

// ------------------------------------------------------------------
extern "C" void kernel_launch(void* const* d_in, const int* in_sizes, int n_in,
                              void* d_out, int out_size, void* d_ws, size_t ws_size,
                              hipStream_t stream) {
  const float* x    = (const float*)d_in[0];
  const float* gate = (const float*)d_in[1];
  const float* up   = (const float*)d_in[2];
  const float* down = (const float*)d_in[3];
  const int*   cnts = (const int*)d_in[4];
  float*       out  = (float*)d_out;

  // workspace: h[TOK * INTR] fp32, then expert offsets
  float* h    = (float*)d_ws;
  int*   offs = (int*)((char*)d_ws + (size_t)TOK * INTR * sizeof(float));

  moe_prefix_kernel<<<dim3(1), dim3(32), 0, stream>>>(cnts, offs);

  dim3 blk(256);
  const size_t ldsA = (size_t)(2 * XSZ + 4 * WSZ) * sizeof(float); // 110592 B
  const size_t ldsB = (size_t)(2 * XSZ + 2 * WSZ) * sizeof(float); //  73728 B
  moe_gateup_kernel<<<dim3(INTR / TN, MAX_MT, NE), blk, ldsA, stream>>>(
      x, gate, up, offs, h);
  moe_down_kernel<<<dim3(HID / TN, MAX_MT, NE), blk, ldsB, stream>>>(
      h, down, offs, out);
}